// AdaptiveSequenceProcessor_16449724745605
// MI455X (gfx1250) — compile-verified
//
#include <hip/hip_runtime.h>
#include <math.h>

typedef __attribute__((ext_vector_type(2))) float v2f;
typedef __attribute__((ext_vector_type(4))) float v4f;
typedef __attribute__((ext_vector_type(8))) float v8f;

#define SEQ 4096
#define BATCH 16
#define DIM 1024
#define HDIM 512
#define LSPAN 32
#define BD (BATCH*DIM)          // 16384
#define CHUNKS 32
#define SPER (SEQ/CHUNKS)       // 128

enum { ACT_NONE = 0, ACT_GELU = 1, ACT_SIGMOID_MUL_A = 2 };

// ---------------- context mean (2-stage) ----------------
// grid(16, 32) x 256 : each thread sums one float4 column over 128 seq steps (NT loads)
__global__ __launch_bounds__(256) void ctx_partial_kernel(const float* __restrict__ features,
                                                          float* __restrict__ partial) {
  int col4 = blockIdx.x * 256 + threadIdx.x;      // 0..4095 (float4 columns)
  int chunk = blockIdx.y;
  const v4f* f = (const v4f*)features;
  v4f s = {0.f, 0.f, 0.f, 0.f};
  long base = (long)chunk * SPER * (BD / 4) + col4;
  for (int i = 0; i < SPER; ++i)
    s += __builtin_nontemporal_load(f + base + (long)i * (BD / 4));
  ((v4f*)partial)[(long)chunk * (BD / 4) + col4] = s;
}

__global__ __launch_bounds__(256) void ctx_final_kernel(const float* __restrict__ partial,
                                                        float* __restrict__ context) {
  int c = blockIdx.x * 256 + threadIdx.x;         // 0..16383
  float s = 0.f;
  for (int ch = 0; ch < CHUNKS; ++ch) s += partial[(long)ch * BD + c];
  context[c] = s * (1.f / SEQ);
}

// ---------------- build comb = [hist, ctx] for both pools ----------------
// grid(1024) x 512 : block = (pool, l, b); 512 float4 = 2048 floats per row
__global__ __launch_bounds__(512) void build_comb_kernel(const float* __restrict__ features,
                                                         const float* __restrict__ context,
                                                         float* __restrict__ comb) {
  int blk = blockIdx.x;
  int pool = blk >> 9;
  int m = blk & 511;                // l*16 + b
  int l = m >> 4, b = m & 15;
  int s = (pool == 0) ? (SEQ - LSPAN + l) : (l * (SEQ / 32));
  const v4f* src0 = (const v4f*)(features + ((long)s * BATCH + b) * DIM);
  const v4f* src1 = (const v4f*)(context + (long)b * DIM);
  v4f* dst = (v4f*)(comb + (long)blk * (2 * DIM));
  int t = threadIdx.x;              // 0..511
  dst[t] = (t < 256) ? src0[t] : src1[t - 256];
}

// ---------------- generic f32 WMMA GEMM: C = act(A @ W^T + bias) ----------------
// A: MxK row-major, W: NxK row-major, C: MxN row-major.
// One wave computes a 16x32 tile with 4 independent WMMA accumulator chains.
template <int ACT>
__global__ __launch_bounds__(32) void wmma_gemm_f32(const float* __restrict__ A,
                                                    const float* __restrict__ W,
                                                    const float* __restrict__ bias,
                                                    float* __restrict__ C,
                                                    int M, int N, int K,
                                                    long strideA, long strideC) {
  int m0 = blockIdx.x * 16;
  int n0 = blockIdx.y * 32;
  A += (long)blockIdx.z * strideA;
  C += (long)blockIdx.z * strideC;
  int lane = threadIdx.x;
  int half = lane >> 4;             // 0: K=0,1  1: K=2,3
  int l16  = lane & 15;
  // A-frag: lane<16 holds A[m0+l16][k..k+1], lane>=16 holds A[m0+l16][k+2..k+3]
  const float* arow = A + (long)(m0 + l16) * K + half * 2;
  // B-frag (B = W^T): lane<16 holds B[k..k+1][n] = W[n][k..k+1]
  const float* w0 = W + (long)(n0 + l16) * K + half * 2;
  const float* w1 = W + (long)(n0 + 16 + l16) * K + half * 2;
  v8f acc0a = {}, acc0b = {}, acc1a = {}, acc1b = {};
  for (int k = 0; k < K; k += 8) {
    v2f a0  = *(const v2f*)(arow + k);
    v2f b00 = *(const v2f*)(w0 + k);
    v2f b10 = *(const v2f*)(w1 + k);
    v2f a1  = *(const v2f*)(arow + k + 4);
    v2f b01 = *(const v2f*)(w0 + k + 4);
    v2f b11 = *(const v2f*)(w1 + k + 4);
    acc0a = __builtin_amdgcn_wmma_f32_16x16x4_f32(false, a0, false, b00, (short)0, acc0a, false, false);
    acc1a = __builtin_amdgcn_wmma_f32_16x16x4_f32(false, a0, false, b10, (short)0, acc1a, false, false);
    acc0b = __builtin_amdgcn_wmma_f32_16x16x4_f32(false, a1, false, b01, (short)0, acc0b, false, false);
    acc1b = __builtin_amdgcn_wmma_f32_16x16x4_f32(false, a1, false, b11, (short)0, acc1b, false, false);
  }
  float bn0 = bias[n0 + l16];
  float bn1 = bias[n0 + 16 + l16];
  for (int r = 0; r < 8; ++r) {
    int m = m0 + r + half * 8;      // C/D layout: VGPR r = row r (lanes<16) / r+8 (lanes>=16)
    float v0 = acc0a[r] + acc0b[r] + bn0;
    float v1 = acc1a[r] + acc1b[r] + bn1;
    if (ACT == ACT_GELU) {
      v0 = 0.5f * v0 * (1.f + erff(v0 * 0.70710678118f));
      v1 = 0.5f * v1 * (1.f + erff(v1 * 0.70710678118f));
    } else if (ACT == ACT_SIGMOID_MUL_A) {
      // fg = A[m,n] * sigmoid(v)   (valid when K == N, A == fused)
      v0 = A[(long)m * K + n0 + l16]      * (1.f / (1.f + __expf(-v0)));
      v1 = A[(long)m * K + n0 + 16 + l16] * (1.f / (1.f + __expf(-v1)));
    }
    C[(long)m * N + n0 + l16]      = v0;
    C[(long)m * N + n0 + 16 + l16] = v1;
  }
}

// ---------------- span head: linear(16x512 -> 16x3) + softmax ----------------
__global__ __launch_bounds__(64) void span_kernel(const float* __restrict__ sph,
                                                  const float* __restrict__ w2,
                                                  const float* __restrict__ b2,
                                                  float* __restrict__ sw) {
  __shared__ float sc[16][3];
  int t = threadIdx.x;
  if (t < 48) {
    int b = t / 3, j = t % 3;
    const float* hr = sph + b * HDIM;
    const float* wr = w2 + j * HDIM;
    float s = b2[j];
    for (int n = 0; n < HDIM; ++n) s += hr[n] * wr[n];
    sc[b][j] = s;
  }
  __syncthreads();
  if (t < 16) {
    float a = sc[t][0], b1 = sc[t][1], c = sc[t][2];
    float mx = fmaxf(a, fmaxf(b1, c));
    float ea = __expf(a - mx), eb = __expf(b1 - mx), ec = __expf(c - mx);
    float inv = 1.f / (ea + eb + ec);
    sw[t * 3 + 0] = ea * inv; sw[t * 3 + 1] = eb * inv; sw[t * 3 + 2] = ec * inv;
  }
}

// ---------------- relevance scores: h(1024x512) . w_rs2 + b ----------------
__global__ __launch_bounds__(256) void scores_kernel(const float* __restrict__ h,
                                                     const float* __restrict__ w2,
                                                     const float* __restrict__ b2,
                                                     float* __restrict__ scores) {
  int idx = blockIdx.x * 256 + threadIdx.x;       // 0..1023 = pool*512 + m
  const float* hr = h + (long)idx * HDIM;
  float s = b2[0];
  for (int n = 0; n < HDIM; ++n) s += hr[n] * w2[n];
  scores[idx] = s;
}

// ---------------- decay softmax over L=32 + weighted pool ----------------
// grid(2,16) x 256
__global__ __launch_bounds__(256) void pool_kernel(const float* __restrict__ scores,
                                                   const float* __restrict__ comb,
                                                   const float* __restrict__ dec_local,
                                                   const float* __restrict__ dec_global,
                                                   float* __restrict__ feat) {
  __shared__ float w[32];
  int pool = blockIdx.x, b = blockIdx.y;
  float decay = (pool == 0) ? dec_local[0] : dec_global[0];
  if (threadIdx.x < 32) {
    int l = threadIdx.x;
    float dv = powf(decay, (float)(31 - l));
    w[l] = scores[pool * 512 + l * 16 + b] + logf(dv + 1e-8f);
  }
  __syncthreads();
  if (threadIdx.x == 0) {
    float mx = w[0];
    for (int l = 1; l < 32; ++l) mx = fmaxf(mx, w[l]);
    float sum = 0.f;
    for (int l = 0; l < 32; ++l) { w[l] = __expf(w[l] - mx); sum += w[l]; }
    float inv = 1.f / sum;
    for (int l = 0; l < 32; ++l) w[l] *= inv;
  }
  __syncthreads();
  for (int d = threadIdx.x; d < DIM; d += 256) {
    float acc = 0.f;
    for (int l = 0; l < 32; ++l)
      acc += w[l] * comb[(((long)pool * 32 + l) * 16 + b) * (2 * DIM) + d];
    feat[((long)pool * 16 + b) * DIM + d] = acc;
  }
}

// ---------------- combined = [local*sw1, global*sw2] ----------------
__global__ __launch_bounds__(256) void combine_kernel(const float* __restrict__ feat,
                                                      const float* __restrict__ sw,
                                                      float* __restrict__ combined) {
  int i = blockIdx.x * 256 + threadIdx.x;         // 0..32767
  int b = i >> 11, d = i & 2047;
  float v;
  if (d < DIM) v = feat[(long)b * DIM + d] * sw[b * 3 + 1];
  else         v = feat[((long)16 + b) * DIM + (d - DIM)] * sw[b * 3 + 2];
  combined[i] = v;
}

// ---------------- LayerNorm over small row set (rows = gridDim.x) ----------------
__global__ __launch_bounds__(256) void ln_rows_kernel(const float* __restrict__ x,
                                                      const float* __restrict__ g,
                                                      const float* __restrict__ bb,
                                                      float* __restrict__ y, int D) {
  __shared__ float red[256];
  int row = blockIdx.x;
  const float* xr = x + (long)row * D;
  float s = 0.f, s2 = 0.f;
  for (int d = threadIdx.x; d < D; d += 256) { float v = xr[d]; s += v; s2 += v * v; }
  red[threadIdx.x] = s; __syncthreads();
  for (int o = 128; o > 0; o >>= 1) { if (threadIdx.x < o) red[threadIdx.x] += red[threadIdx.x + o]; __syncthreads(); }
  float mean = red[0] / D; __syncthreads();
  red[threadIdx.x] = s2; __syncthreads();
  for (int o = 128; o > 0; o >>= 1) { if (threadIdx.x < o) red[threadIdx.x] += red[threadIdx.x + o]; __syncthreads(); }
  float var = red[0] / D - mean * mean;
  float rstd = rsqrtf(var + 1e-5f);
  for (int d = threadIdx.x; d < D; d += 256)
    y[(long)row * D + d] = (xr[d] - mean) * rstd * g[d] + bb[d];
}

// ---------------- out = LN(fg[b] + features[s,b]) : one wave per row ----------------
// NT loads of features, NT stores of output; attention map fused in (lane 0 per wave)
__global__ __launch_bounds__(256) void output_kernel(const float* __restrict__ features,
                                                     const float* __restrict__ fg,
                                                     const float* __restrict__ g,
                                                     const float* __restrict__ bb,
                                                     float* __restrict__ out,
                                                     float* __restrict__ attn) {
  int wave = threadIdx.x >> 5, lane = threadIdx.x & 31;
  long row = (long)blockIdx.x * 8 + wave;          // 0..65535 ; row = s*16+b
  int b = (int)(row & 15);
  const v4f* fr = (const v4f*)(features + row * DIM);
  const v4f* gr = (const v4f*)(fg + (long)b * DIM);
  v4f v[8];
  float s = 0.f, s2 = 0.f;
  for (int i = 0; i < 8; ++i) {
    int idx = i * 32 + lane;
    v4f x = __builtin_nontemporal_load(fr + idx);
    v4f t = x + gr[idx];
    v[i] = t;
    s  += t.x + t.y + t.z + t.w;
    s2 += t.x * t.x + t.y * t.y + t.z * t.z + t.w * t.w;
  }
  for (int o = 16; o > 0; o >>= 1) { s += __shfl_xor(s, o, 32); s2 += __shfl_xor(s2, o, 32); }
  float mean = s * (1.f / DIM);
  float var  = s2 * (1.f / DIM) - mean * mean;
  float rstd = rsqrtf(var + 1e-5f);
  v4f* orow = (v4f*)(out + row * DIM);
  const v4f* g4 = (const v4f*)g;
  const v4f* b4 = (const v4f*)bb;
  for (int i = 0; i < 8; ++i) {
    int idx = i * 32 + lane;
    v4f gg = g4[idx], bv = b4[idx], t = v[i];
    v4f o4 = (t - mean) * rstd * gg + bv;
    __builtin_nontemporal_store(o4, orow + idx);
  }
  if (lane == 0) attn[row] = 1.f / SEQ;
}

// ---------------- workspace layout (float offsets) ----------------
static constexpr long WS_PARTIAL  = 0;                              // 32*16384
static constexpr long WS_CONTEXT  = WS_PARTIAL + (long)CHUNKS * BD; // 16384
static constexpr long WS_COMB     = WS_CONTEXT + BD;                // 2*32*16*2048
static constexpr long WS_H        = WS_COMB + 2L * 32 * 16 * 2048;  // 2*512*512
static constexpr long WS_SCORES   = WS_H + 2L * 512 * 512;          // 1024
static constexpr long WS_FEAT     = WS_SCORES + 1024;               // 2*16*1024
static constexpr long WS_SPH      = WS_FEAT + 2L * 16 * 1024;       // 16*512
static constexpr long WS_SPANW    = WS_SPH + 16L * 512;             // 48 (pad 64)
static constexpr long WS_COMBINED = WS_SPANW + 64;                  // 16*2048
static constexpr long WS_FUSEDPRE = WS_COMBINED + 16L * 2048;       // 16*1024
static constexpr long WS_FUSED    = WS_FUSEDPRE + 16L * 1024;       // 16*1024
static constexpr long WS_FG       = WS_FUSED + 16L * 1024;          // 16*1024

extern "C" void kernel_launch(void* const* d_in, const int* in_sizes, int n_in,
                              void* d_out, int out_size, void* d_ws, size_t ws_size,
                              hipStream_t stream) {
  const float* features = (const float*)d_in[0];
  const float* w_sp1 = (const float*)d_in[1];
  const float* b_sp1 = (const float*)d_in[2];
  const float* w_sp2 = (const float*)d_in[3];
  const float* b_sp2 = (const float*)d_in[4];
  const float* w_rs1 = (const float*)d_in[5];
  const float* b_rs1 = (const float*)d_in[6];
  const float* w_rs2 = (const float*)d_in[7];
  const float* b_rs2 = (const float*)d_in[8];
  const float* decay_local  = (const float*)d_in[9];
  const float* decay_global = (const float*)d_in[10];
  const float* w_ff = (const float*)d_in[11];
  const float* b_ff = (const float*)d_in[12];
  const float* ln_ff_g = (const float*)d_in[13];
  const float* ln_ff_b = (const float*)d_in[14];
  const float* w_gate = (const float*)d_in[15];
  const float* b_gate = (const float*)d_in[16];
  const float* ln_g = (const float*)d_in[17];
  const float* ln_b = (const float*)d_in[18];

  float* ws = (float*)d_ws;
  float* out = (float*)d_out;
  float* attn = out + (long)SEQ * BATCH * DIM;

  // 1) context = mean over S (streams 256 MB, NT)
  ctx_partial_kernel<<<dim3(BD / 1024, CHUNKS), 256, 0, stream>>>(features, ws + WS_PARTIAL);
  ctx_final_kernel<<<BD / 256, 256, 0, stream>>>(ws + WS_PARTIAL, ws + WS_CONTEXT);

  // 2) comb = [hist, ctx] for local+global pools
  build_comb_kernel<<<1024, 512, 0, stream>>>(features, ws + WS_CONTEXT, ws + WS_COMB);

  // 3) span hidden: gelu(context @ w_sp1^T + b_sp1)   (16x1024x512)
  wmma_gemm_f32<ACT_GELU><<<dim3(1, HDIM / 32, 1), 32, 0, stream>>>(
      ws + WS_CONTEXT, w_sp1, b_sp1, ws + WS_SPH, 16, HDIM, DIM, 0, 0);

  // 4) relevance hidden (both pools batched): gelu(comb @ w_rs1^T + b_rs1)  (512x2048x512)
  wmma_gemm_f32<ACT_GELU><<<dim3(512 / 16, HDIM / 32, 2), 32, 0, stream>>>(
      ws + WS_COMB, w_rs1, b_rs1, ws + WS_H, 512, HDIM, 2 * DIM,
      32L * 16 * 2048, 512L * 512);

  // 5) span softmax, relevance scores, decay softmax + pooling
  span_kernel<<<1, 64, 0, stream>>>(ws + WS_SPH, w_sp2, b_sp2, ws + WS_SPANW);
  scores_kernel<<<4, 256, 0, stream>>>(ws + WS_H, w_rs2, b_rs2, ws + WS_SCORES);
  pool_kernel<<<dim3(2, 16), 256, 0, stream>>>(ws + WS_SCORES, ws + WS_COMB,
                                               decay_local, decay_global, ws + WS_FEAT);

  // 6) combined, fused = LN(gelu(combined @ w_ff^T + b_ff))
  combine_kernel<<<(16 * 2048) / 256, 256, 0, stream>>>(ws + WS_FEAT, ws + WS_SPANW,
                                                        ws + WS_COMBINED);
  wmma_gemm_f32<ACT_GELU><<<dim3(1, DIM / 32, 1), 32, 0, stream>>>(
      ws + WS_COMBINED, w_ff, b_ff, ws + WS_FUSEDPRE, 16, DIM, 2 * DIM, 0, 0);
  ln_rows_kernel<<<16, 256, 0, stream>>>(ws + WS_FUSEDPRE, ln_ff_g, ln_ff_b, ws + WS_FUSED, DIM);

  // 7) fg = fused * sigmoid(fused @ w_gate^T + b_gate)  (fused epilogue, K==N)
  wmma_gemm_f32<ACT_SIGMOID_MUL_A><<<dim3(1, DIM / 32, 1), 32, 0, stream>>>(
      ws + WS_FUSED, w_gate, b_gate, ws + WS_FG, 16, DIM, DIM, 0, 0);

  // 8) streaming output: LN(fg[b] + features[s,b]) ; one wave per row; attn fused
  output_kernel<<<(SEQ * BATCH) / 8, 256, 0, stream>>>(features, ws + WS_FG, ln_g, ln_b,
                                                       out, attn);
}